// DLRM_Net_13649406066733
// MI455X (gfx1250) — compile-verified
//
#include <hip/hip_runtime.h>
#include <math.h>

typedef __attribute__((ext_vector_type(2))) float v2f;
typedef __attribute__((ext_vector_type(8))) float v8f;

#define BATCH   4096
#define TTAB    26
#define MSPA    64
#define POOL    32
#define NROWS   200000
#define TSLOTS  32                 // padded feature slots (27 used, 5 zero)
#define TSTRIDE (TSLOTS * MSPA)    // 2048 floats per sample

// ---------------------------------------------------------------------------
// Generic GEMM: C[M][N] = act(A[M][K] * W[N][K]^T + bias[N]), one wave per
// 16x16 C tile, K marched in steps of 4 via V_WMMA_F32_16X16X4_F32.
// K is a compile-time constant: main loop is branch-free; ragged tail (K=13,
// K=415) uses arithmetic masking (clamped index * 0/1) -> no EXEC branches.
// ---------------------------------------------------------------------------
template <int N, int K, int LDC, bool RELU>
__global__ void gemm_wt_relu(const float* __restrict__ A,
                             const float* __restrict__ W,
                             const float* __restrict__ bias,
                             float* __restrict__ C)
{
    constexpr int TILESN = N / 16;
    const int waveId = threadIdx.x >> 5;
    const int lane   = threadIdx.x & 31;

    const int tileFlat = blockIdx.x * (blockDim.x >> 5) + waveId;
    const int tileM = (tileFlat / TILESN) << 4;
    const int tileN = (tileFlat % TILESN) << 4;

    const int half = lane >> 4;      // 0: K pair {0,1}, 1: K pair {2,3}
    const int l    = lane & 15;
    const int m    = tileM + l;      // A row for this lane
    const int n    = tileN + l;      // W row (= C column) for this lane
    const int kOff = half * 2;

    const float* __restrict__ Arow = A + (size_t)m * K;
    const float* __restrict__ Wrow = W + (size_t)n * K;

    constexpr int KAL = K & ~3;      // aligned portion of K
    v8f acc = {};
#pragma unroll 4
    for (int k0 = 0; k0 < KAL; k0 += 4) {
        v2f a, b;
        if constexpr ((K & 1) == 0) {            // rows 8B-aligned -> b64 loads
            a = *(const v2f*)(Arow + k0 + kOff);
            b = *(const v2f*)(Wrow + k0 + kOff);
        } else {
            a.x = Arow[k0 + kOff];
            a.y = Arow[k0 + kOff + 1];
            b.x = Wrow[k0 + kOff];
            b.y = Wrow[k0 + kOff + 1];
        }
        acc = __builtin_amdgcn_wmma_f32_16x16x4_f32(
            false, a, false, b, (short)0, acc, false, false);
    }
    if constexpr (KAL < K) {
        const int ka = KAL + kOff;
        const int kb = ka + 1;
        const float ma = (ka < K) ? 1.0f : 0.0f;   // arithmetic mask
        const float mb = (kb < K) ? 1.0f : 0.0f;
        const int kac = (ka < K) ? ka : (K - 1);   // clamped (always valid) idx
        const int kbc = (kb < K) ? kb : (K - 1);
        v2f a, b;
        a.x = Arow[kac] * ma;
        a.y = Arow[kbc] * mb;
        b.x = Wrow[kac] * ma;
        b.y = Wrow[kbc] * mb;
        acc = __builtin_amdgcn_wmma_f32_16x16x4_f32(
            false, a, false, b, (short)0, acc, false, false);
    }

    // C/D layout: VGPR r, lanes 0-15 -> M = tileM + r; lanes 16-31 -> M + 8.
    const int rowBase = tileM + half * 8;
    const float bv = bias[n];        // column index == n for all lanes
#pragma unroll
    for (int r = 0; r < 8; ++r) {
        float v = acc[r] + bv;
        if (RELU) v = fmaxf(v, 0.0f);
        C[(size_t)(rowBase + r) * LDC + n] = v;
    }
}

// ---------------------------------------------------------------------------
// Zero the 5 padding slots (27..31) of every sample's Tfeat block.
// ---------------------------------------------------------------------------
__global__ void zero_pad_kernel(float* __restrict__ Tfeat)
{
    const int i = blockIdx.x * blockDim.x + threadIdx.x;  // 0 .. 4096*5*64-1
    const int b = i / (5 * MSPA);
    const int r = i % (5 * MSPA);
    Tfeat[(size_t)b * TSTRIDE + (TTAB + 1) * MSPA + r] = 0.0f;
}

// ---------------------------------------------------------------------------
// Embedding bag (sum over POOL rows). One wave per (table, sample).
// Each lane loads one pool index (coalesced), rows are broadcast with
// v_readlane -> SGPR row base; each lane then gathers a float2 slice of the
// 64-wide row (256B coalesced per row). Writes Tfeat[b][1+t][0..63].
// ---------------------------------------------------------------------------
__global__ void emb_pool_kernel(const float* __restrict__ emb,   // 26 x 200000 x 64
                                const int*   __restrict__ lS_i,  // 26 x (4096*32)
                                float* __restrict__ Tfeat)       // 4096 x 32 x 64
{
    const int waveId = threadIdx.x >> 5;
    const int lane   = threadIdx.x & 31;
    const int flat   = blockIdx.x * (blockDim.x >> 5) + waveId;  // 0..26*4096-1
    const int t = flat / BATCH;
    const int b = flat % BATCH;

    const int*   __restrict__ idx = lS_i + (size_t)t * (BATCH * POOL) + (size_t)b * POOL;
    const float* __restrict__ tab = emb + (size_t)t * NROWS * MSPA;

    const int myIdx = idx[lane];     // one coalesced b32 load for all 32 indices

    float ax = 0.0f, ay = 0.0f;
#pragma unroll
    for (int p = 0; p < POOL; ++p) {
        const int row = __builtin_amdgcn_readlane(myIdx, p);   // SGPR row id
        const v2f v = *(const v2f*)(tab + (size_t)row * MSPA + lane * 2);
        ax += v.x;
        ay += v.y;
    }
    float* __restrict__ out = Tfeat + ((size_t)b * TSLOTS + 1 + t) * MSPA + lane * 2;
    out[0] = ax;
    out[1] = ay;
}

// ---------------------------------------------------------------------------
// Feature interaction. One wave per sample: Z = T * T^T with T = 27x64 padded
// to 32 rows (pad rows are zero). A-chunk and B-chunk register layouts
// coincide, so two unconditional b64 loads feed the three needed 16x16 WMMA
// tiles (00, 10, 11). Emits R[b] = [ xbot(64) | strict-lower-tri(Z) (351) ].
// ---------------------------------------------------------------------------
__global__ void interact_kernel(const float* __restrict__ Tfeat, // 4096 x 32 x 64
                                float* __restrict__ R)           // 4096 x 415
{
    const int waveId = threadIdx.x >> 5;
    const int lane   = threadIdx.x & 31;
    const int b = blockIdx.x * (blockDim.x >> 5) + waveId;

    const float* __restrict__ T  = Tfeat + (size_t)b * TSTRIDE;
    float*       __restrict__ Rb = R + (size_t)b * 415;

    const int half = lane >> 4;
    const int l    = lane & 15;
    const int kOff = half * 2;
    const int r1   = 16 + l;                 // padded row 16..31 (>=27 are zero)

    v8f acc00 = {}, acc10 = {}, acc11 = {};
#pragma unroll 4
    for (int k0 = 0; k0 < MSPA; k0 += 4) {
        const int ka = k0 + kOff;
        const v2f v0 = *(const v2f*)(T + l  * MSPA + ka);
        const v2f v1 = *(const v2f*)(T + r1 * MSPA + ka);
        acc00 = __builtin_amdgcn_wmma_f32_16x16x4_f32(
            false, v0, false, v0, (short)0, acc00, false, false);
        acc10 = __builtin_amdgcn_wmma_f32_16x16x4_f32(
            false, v1, false, v0, (short)0, acc10, false, false);
        acc11 = __builtin_amdgcn_wmma_f32_16x16x4_f32(
            false, v1, false, v1, (short)0, acc11, false, false);
    }

    // Dense bottom-MLP features (Tfeat slot 0) -> R[0..63]
    Rb[lane]      = T[lane];
    Rb[lane + 32] = T[lane + 32];

    // Strict lower triangle of Z, flattened as idx(i,j) = i*(i-1)/2 + j, j < i.
#pragma unroll
    for (int r = 0; r < 8; ++r) {
        const int i0 = half * 8 + r;         // rows 0..15 (acc00, j = l)
        if (l < i0)
            Rb[64 + (i0 * (i0 - 1)) / 2 + l] = acc00[r];
        const int i1 = 16 + i0;              // rows 16..31
        if (i1 < TTAB + 1) {
            Rb[64 + (i1 * (i1 - 1)) / 2 + l] = acc10[r];   // j = 0..15 < i1 always
            const int j1 = 16 + l;
            if (j1 < i1)
                Rb[64 + (i1 * (i1 - 1)) / 2 + j1] = acc11[r];
        }
    }
}

// ---------------------------------------------------------------------------
// Final layer: out[b] = sigmoid(dot(t2[b], w) + bias), N=1 -> plain dot.
// ---------------------------------------------------------------------------
__global__ void top_final_kernel(const float* __restrict__ t2,   // 4096 x 256
                                 const float* __restrict__ w,    // 256
                                 const float* __restrict__ bias, // 1
                                 float* __restrict__ out)        // 4096
{
    const int b = blockIdx.x * blockDim.x + threadIdx.x;
    if (b >= BATCH) return;
    const float* __restrict__ row = t2 + (size_t)b * 256;
    float acc = bias[0];
#pragma unroll 8
    for (int k = 0; k < 256; ++k)
        acc = fmaf(row[k], w[k], acc);
    out[b] = 1.0f / (1.0f + expf(-acc));
}

// ---------------------------------------------------------------------------
extern "C" void kernel_launch(void* const* d_in, const int* in_sizes, int n_in,
                              void* d_out, int out_size, void* d_ws, size_t ws_size,
                              hipStream_t stream)
{
    const float* dense_x = (const float*)d_in[0];
    // d_in[1] = lS_o (unused)
    const int*   lS_i    = (const int*)d_in[2];
    const float* emb     = (const float*)d_in[3];
    const float* bw0 = (const float*)d_in[4];
    const float* bb0 = (const float*)d_in[5];
    const float* bw1 = (const float*)d_in[6];
    const float* bb1 = (const float*)d_in[7];
    const float* bw2 = (const float*)d_in[8];
    const float* bb2 = (const float*)d_in[9];
    const float* tw0 = (const float*)d_in[10];
    const float* tb0 = (const float*)d_in[11];
    const float* tw1 = (const float*)d_in[12];
    const float* tb1 = (const float*)d_in[13];
    const float* tw2 = (const float*)d_in[14];
    const float* tb2 = (const float*)d_in[15];
    float* out = (float*)d_out;

    // Workspace layout (floats)
    float* ws    = (float*)d_ws;
    float* Tfeat = ws;                                   // 4096*32*64 = 8,388,608
    float* h1    = Tfeat + (size_t)BATCH * TSTRIDE;      // 4096*512
    float* h2    = h1    + (size_t)BATCH * 512;          // 4096*256
    float* Rbuf  = h2    + (size_t)BATCH * 256;          // 4096*415
    float* t1    = Rbuf  + (size_t)BATCH * 415;          // 4096*512
    float* t2    = t1    + (size_t)BATCH * 512;          // 4096*256

    const int WPB = 4;                 // waves per block for GEMM / interact
    const int THR = WPB * 32;

    // Zero Tfeat padding slots (27..31) for all samples
    {   int n = BATCH * 5 * MSPA;      // 1,310,720
        zero_pad_kernel<<<n / 256, 256, 0, stream>>>(Tfeat); }

    // Bottom MLP
    gemm_wt_relu<512, 13, 512, true><<<(BATCH / 16) * (512 / 16) / WPB, THR, 0, stream>>>(
        dense_x, bw0, bb0, h1);
    gemm_wt_relu<256, 512, 256, true><<<(BATCH / 16) * (256 / 16) / WPB, THR, 0, stream>>>(
        h1, bw1, bb1, h2);
    gemm_wt_relu<64, 256, TSTRIDE, true><<<(BATCH / 16) * (64 / 16) / WPB, THR, 0, stream>>>(
        h2, bw2, bb2, Tfeat);          // writes Tfeat slot 0

    // Embedding bag -> Tfeat slots 1..26   (26*4096 waves, 8 waves/block)
    emb_pool_kernel<<<(TTAB * BATCH) / 8, 8 * 32, 0, stream>>>(emb, lS_i, Tfeat);

    // Pairwise interaction -> R
    interact_kernel<<<BATCH / WPB, THR, 0, stream>>>(Tfeat, Rbuf);

    // Top MLP
    gemm_wt_relu<512, 415, 512, true><<<(BATCH / 16) * (512 / 16) / WPB, THR, 0, stream>>>(
        Rbuf, tw0, tb0, t1);
    gemm_wt_relu<256, 512, 256, true><<<(BATCH / 16) * (256 / 16) / WPB, THR, 0, stream>>>(
        t1, tw1, tb1, t2);

    top_final_kernel<<<BATCH / 256, 256, 0, stream>>>(t2, tw2, tb2, out);
}